// GraphTransformerLayer_89103391522829
// MI455X (gfx1250) — compile-verified
//
#include <hip/hip_runtime.h>
#include <hip/hip_bf16.h>

#define N_NODES 50000
#define N_EDGES 800000
#define IN_F    128
#define HC      128   // HEADS * C = 4 * 32

typedef __attribute__((ext_vector_type(2))) float v2f;
typedef __attribute__((ext_vector_type(4))) float v4f;
typedef __attribute__((ext_vector_type(8))) float v8f;

// ---------------------------------------------------------------------------
// Kernel 1: fused projections  q = xWq+bq, k = xWk+bk, v = xWv+bv,
//           skip = xWskip+bskip (written directly to d_out).
// One block = one 16-row tile of x (N = 3125 * 16 exactly).
// 8 waves/block; wave w owns output columns [16w, 16w+16) for ALL four
// weight matrices, so the A fragment (from LDS) is reused 4x per K-step.
// Uses V_WMMA_F32_16X16X4_F32 (fp32 in/out, K=4), 32 steps over K=128.
// ---------------------------------------------------------------------------
__global__ __launch_bounds__(256) void proj_wmma_kernel(
    const float* __restrict__ x,
    const float* __restrict__ Wq, const float* __restrict__ bq,
    const float* __restrict__ Wk, const float* __restrict__ bk,
    const float* __restrict__ Wv, const float* __restrict__ bv,
    const float* __restrict__ Wsk, const float* __restrict__ bsk,
    float* __restrict__ q, float* __restrict__ k, float* __restrict__ v,
    float* __restrict__ outskip)
{
    __shared__ float xs[16 * IN_F];   // 8 KB x-tile

    const int rowbase = blockIdx.x * 16;

    // Stage the 16x128 fp32 x tile into LDS with float4 loads.
    {
        const v4f* src = reinterpret_cast<const v4f*>(x + (size_t)rowbase * IN_F);
        v4f* dst = reinterpret_cast<v4f*>(xs);
        for (int i = threadIdx.x; i < (16 * IN_F) / 4; i += 256)
            dst[i] = src[i];
    }
    __syncthreads();

    const int lane = threadIdx.x & 31;
    const int wave = threadIdx.x >> 5;
    const int half = lane >> 4;     // 0: lanes 0-15, 1: lanes 16-31
    const int lrow = lane & 15;
    const int col0 = wave * 16;
    const int colb = col0 + lrow;   // this lane's output column

    v8f acc_q = {}; v8f acc_k = {}; v8f acc_v = {}; v8f acc_s = {};

    for (int kk = 0; kk < IN_F; kk += 4) {
        const int krow = kk + 2 * half;
        // A fragment (16x4 fp32): lane holds A[lrow][krow], A[lrow][krow+1]
        v2f a = *reinterpret_cast<const v2f*>(&xs[lrow * IN_F + krow]);
        const int i0 = krow * HC + colb;
        const int i1 = i0 + HC;
        // B fragments (4x16 fp32): lane holds B[krow][colb], B[krow+1][colb]
        v2f bfq; bfq.x = Wq[i0];  bfq.y = Wq[i1];
        v2f bfk; bfk.x = Wk[i0];  bfk.y = Wk[i1];
        v2f bfv; bfv.x = Wv[i0];  bfv.y = Wv[i1];
        v2f bfs; bfs.x = Wsk[i0]; bfs.y = Wsk[i1];

        acc_q = __builtin_amdgcn_wmma_f32_16x16x4_f32(false, a, false, bfq,
                                                      (short)0, acc_q, false, false);
        acc_k = __builtin_amdgcn_wmma_f32_16x16x4_f32(false, a, false, bfk,
                                                      (short)0, acc_k, false, false);
        acc_v = __builtin_amdgcn_wmma_f32_16x16x4_f32(false, a, false, bfv,
                                                      (short)0, acc_v, false, false);
        acc_s = __builtin_amdgcn_wmma_f32_16x16x4_f32(false, a, false, bfs,
                                                      (short)0, acc_s, false, false);
    }

    // C/D layout: VGPR j -> row (j + 8*half), column colb.
    const float biq = bq[colb], bik = bk[colb], biv = bv[colb], bis = bsk[colb];
#pragma unroll
    for (int j = 0; j < 8; ++j) {
        const size_t row = (size_t)(rowbase + j + 8 * half);
        q[row * HC + colb]       = acc_q[j] + biq;
        k[row * HC + colb]       = acc_k[j] + bik;
        v[row * HC + colb]       = acc_v[j] + biv;
        outskip[row * HC + colb] = acc_s[j] + bis;
    }
}

// ---------------------------------------------------------------------------
// Kernel 0: init segment-softmax state: seg_max = -inf, seg_sum = 0.
// ---------------------------------------------------------------------------
__global__ __launch_bounds__(256) void seg_init_kernel(float* __restrict__ seg_max,
                                                       float* __restrict__ seg_sum)
{
    const int i = blockIdx.x * 256 + threadIdx.x;
    if (i < N_NODES * 4) {
        seg_max[i] = __uint_as_float(0xFF800000u);  // -inf
        seg_sum[i] = 0.0f;
    }
}

// Float atomic max via ordered-int-bits trick (valid with -inf init).
__device__ __forceinline__ void atomicMaxFloat(float* addr, float val) {
    if (val >= 0.0f)
        atomicMax(reinterpret_cast<int*>(addr), __float_as_int(val));
    else
        atomicMin(reinterpret_cast<unsigned int*>(addr), __float_as_uint(val));
}

// ---------------------------------------------------------------------------
// Kernel 2: per-edge attention logits + segment max.
// One wave32 per edge; lane l owns channels {l, l+32, l+64, l+96}
// (exactly one channel of each head), 4 simultaneous wave reductions.
// ---------------------------------------------------------------------------
__global__ __launch_bounds__(256) void logits_kernel(
    const int* __restrict__ eidx, const float* __restrict__ q,
    const float* __restrict__ k, float* __restrict__ logits,
    float* __restrict__ seg_max)
{
    const int lane = threadIdx.x & 31;
    const int e = blockIdx.x * 8 + (threadIdx.x >> 5);   // E = 100000*8 exactly
    const int s = eidx[e];
    const int d = eidx[N_EDGES + e];

    const float* qd = q + (size_t)d * HC;
    const float* ks = k + (size_t)s * HC;

    float p0 = qd[lane]      * ks[lane];
    float p1 = qd[lane + 32] * ks[lane + 32];
    float p2 = qd[lane + 64] * ks[lane + 64];
    float p3 = qd[lane + 96] * ks[lane + 96];

#pragma unroll
    for (int off = 16; off > 0; off >>= 1) {
        p0 += __shfl_xor(p0, off, 32);
        p1 += __shfl_xor(p1, off, 32);
        p2 += __shfl_xor(p2, off, 32);
        p3 += __shfl_xor(p3, off, 32);
    }

    const float scale = 0.17677669529663687f;  // 1/sqrt(32)
    if (lane < 4) {
        float lg = (lane == 0 ? p0 : lane == 1 ? p1 : lane == 2 ? p2 : p3) * scale;
        logits[(size_t)e * 4 + lane] = lg;
        atomicMaxFloat(&seg_max[(size_t)d * 4 + lane], lg);
    }
}

// ---------------------------------------------------------------------------
// Kernel 3: expd = exp(logit - seg_max[dst]) in place, segment sum via atomics.
// One thread per (edge, head);  E*4 = 12500*256 exactly.
// ---------------------------------------------------------------------------
__global__ __launch_bounds__(256) void expsum_kernel(
    const int* __restrict__ eidx, float* __restrict__ logits,
    const float* __restrict__ seg_max, float* __restrict__ seg_sum)
{
    const int i = blockIdx.x * 256 + threadIdx.x;
    const int e = i >> 2;
    const int h = i & 3;
    const int d = eidx[N_EDGES + e];
    const float ex = __expf(logits[i] - seg_max[(size_t)d * 4 + h]);
    logits[i] = ex;  // reuse buffer as expd
    atomicAdd(&seg_sum[(size_t)d * 4 + h], ex);
}

// ---------------------------------------------------------------------------
// Kernel 4: scatter  out[dst] += alpha * v[src].  One wave32 per edge,
// lane l handles channels {l, l+32, l+64, l+96} -> 4 f32 atomic adds/lane.
// ---------------------------------------------------------------------------
__global__ __launch_bounds__(256) void scatter_kernel(
    const int* __restrict__ eidx, const float* __restrict__ v,
    const float* __restrict__ expd, const float* __restrict__ seg_sum,
    float* __restrict__ out)
{
    const int lane = threadIdx.x & 31;
    const int e = blockIdx.x * 8 + (threadIdx.x >> 5);
    const int s = eidx[e];
    const int d = eidx[N_EDGES + e];

    const v4f ex = *reinterpret_cast<const v4f*>(expd + (size_t)e * 4);
    const v4f ss = *reinterpret_cast<const v4f*>(seg_sum + (size_t)d * 4);
    const v4f al = ex / (ss + 1e-16f);

    const float* vs = v + (size_t)s * HC;
    float* ob = out + (size_t)d * HC;

    atomicAdd(&ob[lane],      al.x * vs[lane]);
    atomicAdd(&ob[lane + 32], al.y * vs[lane + 32]);
    atomicAdd(&ob[lane + 64], al.z * vs[lane + 64]);
    atomicAdd(&ob[lane + 96], al.w * vs[lane + 96]);
}

// ---------------------------------------------------------------------------
// Kernel 5: LeakyReLU(0.2) in place over d_out.
// ---------------------------------------------------------------------------
__global__ __launch_bounds__(256) void leaky_kernel(float* __restrict__ out)
{
    const int i = blockIdx.x * 256 + threadIdx.x;
    if (i < N_NODES * HC) {
        const float t = out[i];
        out[i] = (t >= 0.0f) ? t : 0.2f * t;
    }
}

// ---------------------------------------------------------------------------
extern "C" void kernel_launch(void* const* d_in, const int* in_sizes, int n_in,
                              void* d_out, int out_size, void* d_ws, size_t ws_size,
                              hipStream_t stream)
{
    const float* x    = (const float*)d_in[0];
    const int*   eidx = (const int*)  d_in[1];
    const float* Wq   = (const float*)d_in[2];
    const float* bq   = (const float*)d_in[3];
    const float* Wk   = (const float*)d_in[4];
    const float* bk   = (const float*)d_in[5];
    const float* Wv   = (const float*)d_in[6];
    const float* bv   = (const float*)d_in[7];
    const float* Wsk  = (const float*)d_in[8];
    const float* bsk  = (const float*)d_in[9];
    float* out = (float*)d_out;

    // Workspace layout (floats): q | k | v | logits/expd | seg_max | seg_sum
    float* ws      = (float*)d_ws;
    float* q       = ws;                               // N*HC = 6,400,000
    float* k       = q  + (size_t)N_NODES * HC;
    float* v       = k  + (size_t)N_NODES * HC;
    float* logits  = v  + (size_t)N_NODES * HC;        // E*4 = 3,200,000
    float* seg_max = logits + (size_t)N_EDGES * 4;     // N*4 = 200,000
    float* seg_sum = seg_max + (size_t)N_NODES * 4;    // N*4 = 200,000

    seg_init_kernel<<<(N_NODES * 4 + 255) / 256, 256, 0, stream>>>(seg_max, seg_sum);

    proj_wmma_kernel<<<N_NODES / 16, 256, 0, stream>>>(
        x, Wq, bq, Wk, bk, Wv, bv, Wsk, bsk, q, k, v, out);

    logits_kernel<<<N_EDGES / 8, 256, 0, stream>>>(eidx, q, k, logits, seg_max);

    expsum_kernel<<<(N_EDGES * 4) / 256, 256, 0, stream>>>(eidx, logits, seg_max, seg_sum);

    scatter_kernel<<<N_EDGES / 8, 256, 0, stream>>>(eidx, v, logits, seg_sum, out);

    leaky_kernel<<<(N_NODES * HC + 255) / 256, 256, 0, stream>>>(out);
}